// CombinedLoss_58497454572164
// MI455X (gfx1250) — compile-verified
//
#include <hip/hip_runtime.h>
#include <stdint.h>

#define NCLS  7
#define BATCH 8
#define HH    512
#define WW    512
#define TILE  32
#define PREG  34   // probs region: tile + radius-1 halo
#define TREG  38   // targets region: tile + radius-3 halo
#define MREG  36   // boundary-mask region: tile + radius-2 halo
#define NPIXF 2097152.0f   // 8*512*512
#define NSLOT 27

typedef float v2f __attribute__((ext_vector_type(2)));
typedef float v8f __attribute__((ext_vector_type(8)));

// acc layout (27 floats in d_ws):
// [0..6]   probsum[c]   (dice union, probs part)
// [7..13]  inter[c]     (dice intersection)
// [14..20] cnt[c]       (onehot counts = dice union onehot part = topo msum)
// [21..25] num[j]       (topo numerators, j = c-2)
// [26]     grand        (focal*0.3/N + boundary*0.2/(N*7), fully folded)

__global__ void zero_acc_kernel(float* __restrict__ acc) {
    int i = threadIdx.x;
    if (i < NSLOT) acc[i] = 0.0f;
}

// Wave32-wide sum of one f32 per lane through the matrix pipe:
// A(16x4) holds lane partials in K=0 slots, B = ones(4x16) =>
// every D column replicates the 16 row-sums; a lane's 8 C/D regs cover
// rows 0-7 (lanes 0-15) or 8-15 (lanes 16-31), so lane n and lane n^16
// together hold the full 32-lane total. Both commit halves to LDS.
__device__ __forceinline__ void wave_wmma_commit(float v, float* slot, int lane) {
    v2f amat; amat[0] = v;    amat[1] = 0.0f;
    v2f bone; bone[0] = 1.0f; bone[1] = 1.0f;
    v8f dmat = {};
    dmat = __builtin_amdgcn_wmma_f32_16x16x4_f32(
        false, amat, false, bone, (short)0, dmat, false, false);
    float part = dmat[0] + dmat[1] + dmat[2] + dmat[3]
               + dmat[4] + dmat[5] + dmat[6] + dmat[7];
    if (lane == 0 || lane == 16) atomicAdd(slot, part);
}

__global__ __launch_bounds__(256) void loss_main_kernel(
    const float* __restrict__ logits,
    const int*   __restrict__ targets,
    const float* __restrict__ cw,
    float*       __restrict__ acc)
{
    __shared__ float s_probs[NCLS][PREG * PREG];  // logits staged, softmaxed in place
    __shared__ int   s_tgt[TREG * TREG];          // targets (-1 = out of image)
    __shared__ int   s_bmask[MREG * MREG];        // per-pixel 7-bit boundary mask
    __shared__ float s_acc[NSLOT];                // workgroup-level accumulators

    const int tid  = threadIdx.x;
    const int lane = tid & 31;
    const int bz   = blockIdx.z;                  // batch
    const int row0 = blockIdx.y * TILE;
    const int col0 = blockIdx.x * TILE;

    if (tid < NSLOT) s_acc[tid] = 0.0f;

    // ---- Phase 1: targets (+3 halo) -> LDS via async copy engine --------
    const int* tgt_b = targets + (size_t)bz * HH * WW;
    for (int i = tid; i < TREG * TREG; i += 256) {
        int r = i / TREG, c = i % TREG;
        int gr = row0 - 3 + r, gc = col0 - 3 + c;
        if (gr >= 0 && gr < HH && gc >= 0 && gc < WW) {
            unsigned long long src =
                (unsigned long long)(uintptr_t)(tgt_b + gr * WW + gc);
            uint32_t dst = (uint32_t)(uintptr_t)&s_tgt[i];
            asm volatile("global_load_async_to_lds_b32 %0, %1, off"
                         :: "v"(dst), "v"(src) : "memory");
        } else {
            s_tgt[i] = -1;   // out-of-image sentinel
        }
    }

    // ---- Phase 2: logits tile (+1 halo) -> LDS via async copy engine ----
    const float* log_b = logits + (size_t)bz * NCLS * HH * WW;
    for (int i = tid; i < NCLS * PREG * PREG; i += 256) {
        int ch  = i / (PREG * PREG);
        int rem = i % (PREG * PREG);
        int r = rem / PREG, c = rem % PREG;
        int gr = row0 - 1 + r, gc = col0 - 1 + c;
        if (gr >= 0 && gr < HH && gc >= 0 && gc < WW) {
            unsigned long long src =
                (unsigned long long)(uintptr_t)(log_b + ((size_t)ch * HH + gr) * WW + gc);
            uint32_t dst = (uint32_t)(uintptr_t)&s_probs[ch][rem];  // low 32 bits = LDS offset
            asm volatile("global_load_async_to_lds_b32 %0, %1, off"
                         :: "v"(dst), "v"(src) : "memory");
        } else {
            s_probs[ch][rem] = 0.0f;  // zero padding (SAME conv)
        }
    }
    asm volatile("s_wait_asynccnt 0x0" ::: "memory");
    __syncthreads();

    // ---- Phase 3a: softmax in place over the 34x34 region ---------------
    for (int i = tid; i < PREG * PREG; i += 256) {
        int r = i / PREG, c = i % PREG;
        int gr = row0 - 1 + r, gc = col0 - 1 + c;
        if (gr < 0 || gr >= HH || gc < 0 || gc >= WW) continue;  // stays zero
        float x[NCLS]; float m = -1e30f;
        #pragma unroll
        for (int k = 0; k < NCLS; ++k) { x[k] = s_probs[k][i]; m = fmaxf(m, x[k]); }
        float s = 0.0f;
        #pragma unroll
        for (int k = 0; k < NCLS; ++k) { x[k] = __expf(x[k] - m); s += x[k]; }
        float inv = 1.0f / s;
        #pragma unroll
        for (int k = 0; k < NCLS; ++k) s_probs[k][i] = x[k] * inv;
    }

    // ---- Phase 3b: boundary bitmask over the 36x36 region ---------------
    // boundary_c(q)=1 iff the 5-point cross is not membership-uniform for c
    //   => mask = allsame ? 0 : union-of-class-bits  (OOB cells contribute 0)
    for (int i = tid; i < MREG * MREG; i += 256) {
        int r = i / MREG, c = i % MREG;
        int gr = row0 - 2 + r, gc = col0 - 2 + c;
        int bm = 0;
        if (gr >= 0 && gr < HH && gc >= 0 && gc < WW) {
            int ti = (r + 1) * TREG + (c + 1);  // mask->tgt coord shift
            int t0 = s_tgt[ti];
            int tN = s_tgt[ti - TREG], tS = s_tgt[ti + TREG];
            int tWv = s_tgt[ti - 1],   tE = s_tgt[ti + 1];
            int b0 = 1 << t0;
            int b1 = (tN  >= 0) ? (1 << tN)  : 0;
            int b2 = (tS  >= 0) ? (1 << tS)  : 0;
            int b3 = (tWv >= 0) ? (1 << tWv) : 0;
            int b4 = (tE  >= 0) ? (1 << tE)  : 0;
            int present = b0 | b1 | b2 | b3 | b4;
            bool allsame = (b0 == b1) && (b0 == b2) && (b0 == b3) && (b0 == b4);
            bm = allsame ? 0 : present;
        }
        s_bmask[i] = bm;
    }
    __syncthreads();

    // ---- Phase 4: per-pixel loss terms (4 interior pixels / thread) -----
    float cwr[NCLS];
    #pragma unroll
    for (int k = 0; k < NCLS; ++k) cwr[k] = cw[k];   // uniform scalar loads

    float grand = 0.0f;
    float psum[NCLS] = {}, inter[NCLS] = {}, cnt[NCLS] = {}, num[5] = {};

    #pragma unroll
    for (int k4 = 0; k4 < 4; ++k4) {
        int idx = tid + k4 * 256;
        int r = idx >> 5, c = idx & 31;

        int t = s_tgt[(r + 3) * TREG + (c + 3)];
        int pidx = (r + 1) * PREG + (c + 1);
        float p[NCLS];
        #pragma unroll
        for (int k = 0; k < NCLS; ++k) p[k] = s_probs[k][pidx];

        // focal
        float pt_prob = 0.0f, wsel = 0.0f;
        #pragma unroll
        for (int k = 0; k < NCLS; ++k) {
            bool s = (k == t);
            pt_prob = s ? p[k] : pt_prob;
            wsel    = s ? cwr[k] : wsel;
        }
        float ce  = wsel * (-__logf(pt_prob));
        float ptv = __expf(-ce);
        float om  = 1.0f - ptv;
        float focal = om * om * ce;

        // dice accumulators
        #pragma unroll
        for (int k = 0; k < NCLS; ++k) {
            psum[k] += p[k];
            bool s = (k == t);
            inter[k] += s ? p[k] : 0.0f;
            cnt[k]   += s ? 1.0f : 0.0f;
        }

        // boundary-weighted BCE: region mask = OR of 5x5 boundary bitmasks
        int rm = 0;
        #pragma unroll
        for (int dr = 0; dr < 5; ++dr)
            #pragma unroll
            for (int dc = 0; dc < 5; ++dc)
                rm |= s_bmask[(r + dr) * MREG + (c + dc)];
        float bsum = 0.0f;
        #pragma unroll
        for (int k = 0; k < NCLS; ++k) {
            float bw = ((rm >> k) & 1) ? 5.0f : 1.0f;
            float lg = (k == t) ? __logf(p[k] + 1e-7f) : __logf(1.0f - p[k] + 1e-7f);
            bsum -= bw * lg;
        }

        // topo (only classes 2..6; mask = onehot at center => only class t)
        if (t >= 2) {
            const float* plane = &s_probs[0][0] + t * (PREG * PREG);
            float pcs = 0.0f, tcs = 0.0f;
            int pr = r + 1, pcc = c + 1;
            int tr = r + 3, tcc = c + 3;
            #pragma unroll
            for (int dr = -1; dr <= 1; ++dr)
                #pragma unroll
                for (int dc = -1; dc <= 1; ++dc) {
                    pcs += plane[(pr + dr) * PREG + (pcc + dc)];
                    tcs += (s_tgt[(tr + dr) * TREG + (tcc + dc)] == t) ? 1.0f : 0.0f;
                }
            float diff = fabsf(pcs - tcs) * (1.0f / 9.0f);
            #pragma unroll
            for (int j = 0; j < 5; ++j) num[j] += (t == j + 2) ? diff : 0.0f;
        }

        grand += (0.3f / NPIXF) * focal + (0.2f / (NPIXF * 7.0f)) * bsum;
    }

    // ---- Phase 5: wave-level reduction via matrix pipe, then LDS --------
    // One WMMA per accumulator slot collapses all 32 lane partials; only
    // lanes 0 and 16 touch LDS (2 atomics/slot/wave vs 32 before).
    #pragma unroll
    for (int k = 0; k < NCLS; ++k) wave_wmma_commit(psum[k],  &s_acc[k],      lane);
    #pragma unroll
    for (int k = 0; k < NCLS; ++k) wave_wmma_commit(inter[k], &s_acc[7 + k],  lane);
    #pragma unroll
    for (int k = 0; k < NCLS; ++k) wave_wmma_commit(cnt[k],   &s_acc[14 + k], lane);
    #pragma unroll
    for (int j = 0; j < 5; ++j)    wave_wmma_commit(num[j],   &s_acc[21 + j], lane);
    wave_wmma_commit(grand, &s_acc[26], lane);

    __syncthreads();
    if (tid < NSLOT) atomicAdd(&acc[tid], s_acc[tid]);
}

__global__ void loss_finalize_kernel(const float* __restrict__ acc,
                                     float* __restrict__ out) {
    if (threadIdx.x == 0) {
        float dice = 0.0f;
        #pragma unroll
        for (int c = 0; c < NCLS; ++c) {
            float inter = acc[7 + c];
            float uni   = acc[c] + acc[14 + c];
            dice += 1.0f - (2.0f * inter + 1e-6f) / (uni + 1e-6f);
        }
        dice *= (1.0f / 7.0f);
        float topo = 0.0f;
        #pragma unroll
        for (int j = 0; j < 5; ++j) {
            float msum = acc[14 + 2 + j];
            float nm   = acc[21 + j];
            topo += (msum >= 1.0f) ? (nm / fmaxf(msum, 1.0f)) : 0.0f;
        }
        topo *= (1.0f / 5.0f);
        out[0] = acc[26] + 0.3f * dice + 0.2f * topo;
    }
}

extern "C" void kernel_launch(void* const* d_in, const int* in_sizes, int n_in,
                              void* d_out, int out_size, void* d_ws, size_t ws_size,
                              hipStream_t stream) {
    const float* logits  = (const float*)d_in[0];
    const int*   targets = (const int*)d_in[1];
    const float* cw      = (const float*)d_in[2];
    float* acc = (float*)d_ws;
    float* out = (float*)d_out;

    hipLaunchKernelGGL(zero_acc_kernel, dim3(1), dim3(32), 0, stream, acc);
    dim3 grid(WW / TILE, HH / TILE, BATCH);   // 16 x 16 x 8 tiles
    hipLaunchKernelGGL(loss_main_kernel, grid, dim3(256), 0, stream,
                       logits, targets, cw, acc);
    hipLaunchKernelGGL(loss_finalize_kernel, dim3(1), dim3(32), 0, stream, acc, out);
}